// NonLocal1d_21766894256878
// MI455X (gfx1250) — compile-verified
//
#include <hip/hip_runtime.h>
#include <math.h>

#define C_DIM 64
#define INTER 32
#define BATCH 4
#define LSEQ  4096

typedef __attribute__((ext_vector_type(2))) float v2f;
typedef __attribute__((ext_vector_type(8))) float v8f;

// D = A(16x4) * B(4x16) + C, fp32 WMMA
__device__ __forceinline__ v8f wmma4(v2f a, v2f b, v8f c) {
  return __builtin_amdgcn_wmma_f32_16x16x4_f32(
      /*neg_a=*/false, a, /*neg_b=*/false, b,
      /*c_mod=*/(short)0, c, /*reuse_a=*/false, /*reuse_b=*/false);
}

// Swap the two 16-lane halves of the wave (lane i <-> lane i^16) in the VALU
// pipe: v_permlanex16_b32 with identity nibble selectors. Avoids the
// ds_bpermute + s_wait_dscnt round-trip that __shfl_xor(v,16) generates.
__device__ __forceinline__ float xswap16(float v) {
  int i = __builtin_bit_cast(int, v);
  int r = __builtin_amdgcn_permlanex16(i, i, 0x76543210, 0xfedcba98, false, false);
  return __builtin_bit_cast(float, r);
}

// Given V[0..7] in C-matrix layout (lane t holds X[r + 8*hi][col], col=t&15,
// hi=t>>4), produce the B-fragment for K-chunk k (k=0..3):
//   b.x = X[4k + 2*hi][col] ; b.y = X[4k + 2*hi + 1][col]
__device__ __forceinline__ v2f bfrag16(const float* V, int k, int hi) {
  v2f r;
  if (k < 2) {
    float ox = xswap16(V[4 * k + 2]);
    float oy = xswap16(V[4 * k + 3]);
    r.x = hi ? ox : V[4 * k + 0];
    r.y = hi ? oy : V[4 * k + 1];
  } else {
    float ox = xswap16(V[4 * k - 8]);
    float oy = xswap16(V[4 * k - 7]);
    r.x = hi ? V[4 * k - 6] : ox;
    r.y = hi ? V[4 * k - 5] : oy;
  }
  return r;
}

// ---------------------------------------------------------------------------
// Stage 1: g_x / theta_x / phi_x = W[32,64] @ x[b,64,L] + bias  (tiny GEMMs)
// ---------------------------------------------------------------------------
__global__ void __launch_bounds__(256)
proj_kernel(const float* __restrict__ x,
            const float* __restrict__ g_w, const float* __restrict__ g_b,
            const float* __restrict__ th_w, const float* __restrict__ th_b,
            const float* __restrict__ ph_w, const float* __restrict__ ph_b,
            float* __restrict__ g_x, float* __restrict__ th_x,
            float* __restrict__ ph_x) {
  int tid = blockIdx.x * blockDim.x + threadIdx.x;
  int l = tid & (LSEQ - 1);
  int o = (tid >> 12) & (INTER - 1);
  int b = tid >> 17;
  const float* xp = x + (size_t)b * C_DIM * LSEQ + l;
  float sg = 0.f, st = 0.f, sp = 0.f;
#pragma unroll 8
  for (int c = 0; c < C_DIM; ++c) {
    float xv = xp[(size_t)c * LSEQ];
    sg = fmaf(g_w[o * C_DIM + c], xv, sg);
    st = fmaf(th_w[o * C_DIM + c], xv, st);
    sp = fmaf(ph_w[o * C_DIM + c], xv, sp);
  }
  g_x[tid] = sg + g_b[o];
  th_x[tid] = st + th_b[o];
  ph_x[tid] = sp + ph_b[o];
}

// ---------------------------------------------------------------------------
// Stage 2: fused flash attention + output projection + residual.
// Each 16-query tile is handled by a PAIR of waves, each covering half of the
// 256 key tiles with its own online-softmax state; partials merged via LDS
// (standard flash merge). Doubles resident waves (2048) for latency hiding.
//   S'[m][l] = phi^T * theta          (8x wmma f32 16x16x4, K=o=32)
//   P' = exp(S' - rowmax)             (softmax over m == per-lane column l)
//   acc[o][l] += g * P'               (8x wmma, two o-halves)
// Epilogue (even wave): out[c][l] = W_w*y + W_b + x  (32x wmma), fused.
// ---------------------------------------------------------------------------
__global__ void __launch_bounds__(256)
attn_kernel(const float* __restrict__ th_x, const float* __restrict__ ph_x,
            const float* __restrict__ g_x, const float* __restrict__ W_w,
            const float* __restrict__ W_b, const float* __restrict__ x,
            float* __restrict__ out) {
  __shared__ float lds_m[8][32];
  __shared__ float lds_l[8][32];
  __shared__ float lds_acc[4][16][32];

  const int lane = threadIdx.x & 31;
  const int wave = threadIdx.x >> 5;
  const int col = lane & 15;
  const int hi = lane >> 4;
  const int half = wave & 1;   // which half of the key range
  const int pair = wave >> 1;  // query tile within block

  const int qt = blockIdx.x * 4 + pair;  // 0..1023 query tiles
  const int b = qt >> 8;                 // 256 tiles per batch
  const int l0 = (qt & 255) << 4;

  const size_t base = (size_t)b * INTER * LSEQ;
  const float* th = th_x + base;
  const float* ph = ph_x + base;
  const float* gg = g_x + base;

  // theta B-fragments (K = o), reused for all key tiles:
  v2f btheta[8];
#pragma unroll
  for (int k = 0; k < 8; ++k) {
    int o = 4 * k + 2 * hi;
    btheta[k].x = th[(size_t)o * LSEQ + l0 + col];
    btheta[k].y = th[(size_t)(o + 1) * LSEQ + l0 + col];
  }

  v8f acc0 = {};  // y rows o = 0..15   (C-layout, N = l)
  v8f acc1 = {};  // y rows o = 16..31
  float m_run = -INFINITY;
  float l_run = 0.f;

  const int mt_beg = half * (LSEQ / 32);  // 128 key tiles per wave
  const int mt_end = mt_beg + (LSEQ / 32);
  for (int mt = mt_beg; mt < mt_end; ++mt) {
    const int m0 = mt << 4;

    // phi^T A-fragments (M = m, K = o): a.x = phi[4k+2hi][m0+col]
    v2f aphi[8];
#pragma unroll
    for (int k = 0; k < 8; ++k) {
      int o = 4 * k + 2 * hi;
      aphi[k].x = ph[(size_t)o * LSEQ + m0 + col];
      aphi[k].y = ph[(size_t)(o + 1) * LSEQ + m0 + col];
    }
    // g A-fragments (M = o, K = m), two o-halves; m-pair contiguous -> b64
    v2f ag0[4], ag1[4];
#pragma unroll
    for (int k = 0; k < 4; ++k) {
      int mm = m0 + 4 * k + 2 * hi;
      ag0[k] = *(const v2f*)(gg + (size_t)col * LSEQ + mm);
      ag1[k] = *(const v2f*)(gg + (size_t)(col + 16) * LSEQ + mm);
    }

    // scores S'[m][l], K = 32 in 8 chunks of 4
    v8f S = {};
#pragma unroll
    for (int k = 0; k < 8; ++k) S = wmma4(aphi[k], btheta[k], S);

    // online softmax over m for each column l (lane-local + one half-swap)
    float P[8];
#pragma unroll
    for (int r = 0; r < 8; ++r) P[r] = S[r];
    float tmax = P[0];
#pragma unroll
    for (int r = 1; r < 8; ++r) tmax = fmaxf(tmax, P[r]);
    tmax = fmaxf(tmax, xswap16(tmax));
    float m_new = fmaxf(m_run, tmax);
    float scale = __expf(m_run - m_new);  // first iter: exp(-inf) = 0
    float tsum = 0.f;
#pragma unroll
    for (int r = 0; r < 8; ++r) {
      P[r] = __expf(P[r] - m_new);
      tsum += P[r];
    }
    tsum += xswap16(tsum);
    l_run = l_run * scale + tsum;
    m_run = m_new;
#pragma unroll
    for (int r = 0; r < 8; ++r) {
      acc0[r] *= scale;
      acc1[r] *= scale;
    }

    // acc[o][l] += g[o][m] * P'[m][l]
#pragma unroll
    for (int k = 0; k < 4; ++k) {
      v2f bp = bfrag16(P, k, hi);
      acc0 = wmma4(ag0[k], bp, acc0);
      acc1 = wmma4(ag1[k], bp, acc1);
    }
  }

  // -------- flash merge of the wave pair --------
  lds_m[wave][lane] = m_run;
  lds_l[wave][lane] = l_run;
  __syncthreads();
  float m_p = lds_m[wave ^ 1][lane];
  float l_p = lds_l[wave ^ 1][lane];
  float M = fmaxf(m_run, m_p);
  float s = __expf(m_run - M);
  float s_p = __expf(m_p - M);
  float Ltot = l_run * s + l_p * s_p;
  if (half) {  // odd wave: publish rescaled partial accumulator
#pragma unroll
    for (int r = 0; r < 8; ++r) {
      lds_acc[pair][r][lane] = acc0[r] * s;
      lds_acc[pair][8 + r][lane] = acc1[r] * s;
    }
  }
  __syncthreads();
  if (half) return;  // even wave finishes the tile

  float rinv = 1.0f / Ltot;
  float Y0[8], Y1[8];
#pragma unroll
  for (int r = 0; r < 8; ++r) {
    Y0[r] = (acc0[r] * s + lds_acc[pair][r][lane]) * rinv;
    Y1[r] = (acc1[r] * s + lds_acc[pair][8 + r][lane]) * rinv;
  }

  // Re-fragment y (C-layout -> B-fragments over K = o = 32)
  v2f by[8];
#pragma unroll
  for (int k = 0; k < 4; ++k) {
    by[k] = bfrag16(Y0, k, hi);
    by[k + 4] = bfrag16(Y1, k, hi);
  }

  // out[c][l] = W_w[c][o] * y[o][l] + W_b[c] + x[b][c][l]
  const float* xb = x + (size_t)b * C_DIM * LSEQ;
  float* ob = out + (size_t)b * C_DIM * LSEQ;
#pragma unroll
  for (int ct = 0; ct < 4; ++ct) {
    v8f O = {};
#pragma unroll
    for (int k = 0; k < 8; ++k) {
      v2f aw = *(const v2f*)(W_w + (size_t)(ct * 16 + col) * INTER + 4 * k + 2 * hi);
      O = wmma4(aw, by[k], O);
    }
#pragma unroll
    for (int r = 0; r < 8; ++r) {
      int c = ct * 16 + r + 8 * hi;
      size_t idx = (size_t)c * LSEQ + l0 + col;
      ob[idx] = O[r] + W_b[c] + xb[idx];
    }
  }
}

extern "C" void kernel_launch(void* const* d_in, const int* in_sizes, int n_in,
                              void* d_out, int out_size, void* d_ws, size_t ws_size,
                              hipStream_t stream) {
  const float* x = (const float*)d_in[0];
  const float* g_w = (const float*)d_in[1];
  const float* g_b = (const float*)d_in[2];
  const float* th_w = (const float*)d_in[3];
  const float* th_b = (const float*)d_in[4];
  const float* ph_w = (const float*)d_in[5];
  const float* ph_b = (const float*)d_in[6];
  const float* W_w = (const float*)d_in[7];
  const float* W_b = (const float*)d_in[8];
  float* out = (float*)d_out;

  float* ws = (float*)d_ws;
  const size_t n = (size_t)BATCH * INTER * LSEQ;  // 524288 floats each
  float* g_x = ws;
  float* th_x = ws + n;
  float* ph_x = ws + 2 * n;

  proj_kernel<<<(BATCH * INTER * LSEQ) / 256, 256, 0, stream>>>(
      x, g_w, g_b, th_w, th_b, ph_w, ph_b, g_x, th_x, ph_x);

  // 1024 query tiles x 2 key-halves = 2048 waves; 8 waves per block
  attn_kernel<<<(BATCH * (LSEQ / 16) * 2) / 8, 256, 0, stream>>>(
      th_x, ph_x, g_x, W_w, W_b, x, out);
}